// EfficientGRAMEncoder_42202348651215
// MI455X (gfx1250) — compile-verified
//
#include <hip/hip_runtime.h>
#include <math.h>

#define HDIM 128
#define BN_EPS 1e-5f

typedef __attribute__((ext_vector_type(2))) float v2f;
typedef __attribute__((ext_vector_type(8))) float v8f;

// ---------------------------------------------------------------------------
// GCN normalization precompute: deg (with self loop), dinv = rsqrt(deg),
// per-edge norm = dinv[src]*dinv[dst].  Computed ONCE, reused by all 4 layers.
// ---------------------------------------------------------------------------
__global__ __launch_bounds__(256) void k_deg_init(float* __restrict__ deg, int N) {
  int i = blockIdx.x * 256 + threadIdx.x;
  if (i < N) deg[i] = 1.0f;  // self-loop contribution
}

__global__ __launch_bounds__(256) void k_deg_accum(const int* __restrict__ dst,
                                                   float* __restrict__ deg, int E) {
  int e = blockIdx.x * 256 + threadIdx.x;
  if (e < E) atomicAdd(&deg[dst[e]], 1.0f);
}

__global__ __launch_bounds__(256) void k_dinv(float* __restrict__ deg, int N) {
  int i = blockIdx.x * 256 + threadIdx.x;
  if (i < N) {
    float d = deg[i];
    deg[i] = (d > 0.0f) ? rsqrtf(d) : 0.0f;
  }
}

__global__ __launch_bounds__(256) void k_norm(const int* __restrict__ src,
                                              const int* __restrict__ dst,
                                              const float* __restrict__ dinv,
                                              float* __restrict__ normw, int E) {
  int e = blockIdx.x * 256 + threadIdx.x;
  if (e < E) normw[e] = dinv[src[e]] * dinv[dst[e]];
}

// ---------------------------------------------------------------------------
// GEMM: Y[N x 128] = X[N x 128] @ W[128 x 128]  via V_WMMA_F32_16X16X4_F32.
// Block = 256 threads = 8 waves. Block computes a 16-row x 128-col slab.
// A tile (16x128) is staged in LDS (stride 132 to dodge bank conflicts) and
// shared by all 8 waves; wave w owns columns [16w,16w+16). B is read from W,
// which is 64KB, reused 3125x, and stays WGP$/L2 resident.
// ---------------------------------------------------------------------------
__global__ __launch_bounds__(256) void k_gemm_wmma(const float* __restrict__ X,
                                                   const float* __restrict__ W,
                                                   float* __restrict__ Y, int N) {
  __shared__ float As[16][132];
  const int tid  = threadIdx.x;
  const int wave = tid >> 5;
  const int lane = tid & 31;
  const int row0 = blockIdx.x * 16;

  // Cooperative load of the 16x128 A tile (zero-padded past N).
  for (int i = tid; i < 16 * (HDIM / 4); i += 256) {
    int r  = i >> 5;      // HDIM/4 == 32 float4 per row
    int c4 = i & 31;
    float4 v = make_float4(0.f, 0.f, 0.f, 0.f);
    int gr = row0 + r;
    if (gr < N) v = *(const float4*)(X + (size_t)gr * HDIM + c4 * 4);
    *(float4*)&As[r][c4 * 4] = v;
  }
  __syncthreads();

  // ISA fragment layout (32-bit A 16x4): lanes 0-15 hold K={0,1}, lanes 16-31
  // hold K={2,3}; lane&15 selects row (A) / col (B).
  const int col   = wave * 16 + (lane & 15);
  const int khalf = (lane >> 4) << 1;   // 0 or 2
  const int arow  = lane & 15;

  v8f acc = {0.f, 0.f, 0.f, 0.f, 0.f, 0.f, 0.f, 0.f};
#pragma unroll
  for (int k0 = 0; k0 < HDIM; k0 += 4) {
    v2f a;
    a.x = As[arow][k0 + khalf + 0];
    a.y = As[arow][k0 + khalf + 1];
    v2f b;
    b.x = W[(size_t)(k0 + khalf + 0) * HDIM + col];
    b.y = W[(size_t)(k0 + khalf + 1) * HDIM + col];
    // 8 args: (neg_a, A, neg_b, B, c_mod, C, reuse_a, reuse_b)
    acc = __builtin_amdgcn_wmma_f32_16x16x4_f32(
        false, a, false, b, (short)0, acc, false, false);
  }

  // C/D layout: VGPR v holds row (v + 8*(lane>>4)), col = lane&15 in tile.
  const int mbase = (lane >> 4) << 3;  // 0 or 8
#pragma unroll
  for (int v = 0; v < 8; ++v) {
    int gr = row0 + mbase + v;
    if (gr < N) Y[(size_t)gr * HDIM + col] = acc[v];
  }
}

// ---------------------------------------------------------------------------
// Hout = bias + dinv[i]^2 * Y[i]   (bias + self-loop term, no atomics needed)
// ---------------------------------------------------------------------------
__global__ __launch_bounds__(256) void k_init_self(const float* __restrict__ Y,
                                                   const float* __restrict__ dinv,
                                                   const float* __restrict__ bias,
                                                   float* __restrict__ Hout, int N) {
  int i = blockIdx.x * 256 + threadIdx.x;   // over N * 32 float4 elements
  if (i >= N * (HDIM / 4)) return;
  int row = i >> 5;
  int c4  = i & 31;
  float di = dinv[row];
  float w  = di * di;
  const float4 y = *(const float4*)(Y + (size_t)row * HDIM + c4 * 4);
  const float4 b = *(const float4*)(bias + c4 * 4);
  float4 o = make_float4(b.x + y.x * w, b.y + y.y * w, b.z + y.z * w, b.w + y.w * w);
  *(float4*)(Hout + (size_t)row * HDIM + c4 * 4) = o;
}

// ---------------------------------------------------------------------------
// Edge aggregation: one wave per edge, lane owns 4 channels (128b gather,
// 4x global_atomic_add_f32 scatter). All rows are L2-resident on MI455X
// (Y + Hout = 51MB << 192MB L2), so the random traffic stays on-die.
// ---------------------------------------------------------------------------
__global__ __launch_bounds__(256) void k_agg(const float* __restrict__ Y,
                                             const int* __restrict__ src,
                                             const int* __restrict__ dst,
                                             const float* __restrict__ normw,
                                             float* __restrict__ Hout, int E) {
  int e = blockIdx.x * 8 + (threadIdx.x >> 5);
  if (e >= E) return;
  int lane = threadIdx.x & 31;
  int s = src[e];
  int d = dst[e];
  float w = normw[e];
  const float4 y = *(const float4*)(Y + (size_t)s * HDIM + lane * 4);
  float* out = Hout + (size_t)d * HDIM + lane * 4;
  atomicAdd(out + 0, y.x * w);
  atomicAdd(out + 1, y.y * w);
  atomicAdd(out + 2, y.z * w);
  atomicAdd(out + 3, y.w * w);
}

// ---------------------------------------------------------------------------
// BatchNorm statistics: per-block LDS partial sums, one global atomic per
// block per channel.  stats[0:H) = sum, stats[H:2H) = sum of squares.
// ---------------------------------------------------------------------------
__global__ __launch_bounds__(256) void k_zero_stats(float* __restrict__ stats) {
  int i = threadIdx.x;
  if (i < 2 * HDIM) stats[i] = 0.0f;
}

__global__ __launch_bounds__(256) void k_stats(const float* __restrict__ Hv,
                                               float* __restrict__ stats, int N) {
  __shared__ float ssum[HDIM];
  __shared__ float ssq[HDIM];
  int tid = threadIdx.x;
  if (tid < HDIM) { ssum[tid] = 0.0f; ssq[tid] = 0.0f; }
  __syncthreads();
  int c  = tid & (HDIM - 1);
  int rs = tid >> 7;                 // two rows processed in parallel
  int row0   = blockIdx.x * 256;
  int rowEnd = min(row0 + 256, N);
  float ps = 0.0f, pq = 0.0f;
  for (int r = row0 + rs; r < rowEnd; r += 2) {
    float v = Hv[(size_t)r * HDIM + c];
    ps += v;
    pq += v * v;
  }
  atomicAdd(&ssum[c], ps);
  atomicAdd(&ssq[c], pq);
  __syncthreads();
  if (tid < HDIM) {
    atomicAdd(&stats[tid], ssum[tid]);
    atomicAdd(&stats[HDIM + tid], ssq[tid]);
  }
}

// Fold mean/var/gamma/beta into per-channel scale+shift.
__global__ void k_finalize(float* __restrict__ stats,
                           const float* __restrict__ gamma,
                           const float* __restrict__ beta, int N) {
  int c = threadIdx.x;
  if (c < HDIM) {
    float s  = stats[c];
    float q  = stats[HDIM + c];
    float mu = s / (float)N;
    float var = q / (float)N - mu * mu;   // biased var, matches reference
    float sc = gamma[c] * rsqrtf(var + BN_EPS);
    stats[c]        = sc;
    stats[HDIM + c] = beta[c] - mu * sc;
  }
}

// ---------------------------------------------------------------------------
// BN apply + exact GELU (+ optional attn scale on the last layer).
// ---------------------------------------------------------------------------
__global__ __launch_bounds__(256) void k_apply(const float* __restrict__ Hv,
                                               const float* __restrict__ stats,
                                               const float* __restrict__ attn,
                                               float* __restrict__ out, int N) {
  int i = blockIdx.x * 256 + threadIdx.x;
  if (i >= N * (HDIM / 4)) return;
  int row = i >> 5;
  int c4  = i & 31;
  const float4 h  = *(const float4*)(Hv + (size_t)row * HDIM + c4 * 4);
  const float4 sc = *(const float4*)(stats + c4 * 4);
  const float4 sh = *(const float4*)(stats + HDIM + c4 * 4);
  float t[4] = {h.x * sc.x + sh.x, h.y * sc.y + sh.y,
                h.z * sc.z + sh.z, h.w * sc.w + sh.w};
  float g[4];
#pragma unroll
  for (int j = 0; j < 4; ++j) {
    float x = t[j];
    g[j] = 0.5f * x * (1.0f + erff(x * 0.70710678118654752f));  // exact GELU
  }
  if (attn) {
    const float4 a = *(const float4*)(attn + c4 * 4);
    g[0] *= a.x; g[1] *= a.y; g[2] *= a.z; g[3] *= a.w;
  }
  float4 o = make_float4(g[0], g[1], g[2], g[3]);
  *(float4*)(out + (size_t)row * HDIM + c4 * 4) = o;
}

// ---------------------------------------------------------------------------
// Host-side orchestration.
// Workspace layout (floats): bufA[N*H] | Ybuf[N*H] | dinv[N] | normw[E] |
// stats[2H]  (~55 MB).  d_out doubles as the aggregation buffer each layer;
// the last layer's BN+GELU+attn runs in place on d_out.
// ---------------------------------------------------------------------------
extern "C" void kernel_launch(void* const* d_in, const int* in_sizes, int n_in,
                              void* d_out, int out_size, void* d_ws, size_t ws_size,
                              hipStream_t stream) {
  const float* x      = (const float*)d_in[0];
  const int*   ei     = (const int*)d_in[1];
  const float* Ws     = (const float*)d_in[2];
  const float* bs     = (const float*)d_in[3];
  const float* gammas = (const float*)d_in[4];
  const float* betas  = (const float*)d_in[5];
  const float* attn   = (const float*)d_in[6];

  const int N = in_sizes[0] / HDIM;
  const int E = in_sizes[1] / 2;
  const int L = in_sizes[2] / (HDIM * HDIM);
  const int* src = ei;       // edge_index[0]
  const int* dst = ei + E;   // edge_index[1]

  float* ws    = (float*)d_ws;
  float* bufA  = ws;
  float* Ybuf  = bufA + (size_t)N * HDIM;
  float* dinv  = Ybuf + (size_t)N * HDIM;
  float* normw = dinv + N;
  float* stats = normw + E;
  float* Hagg  = (float*)d_out;

  dim3 b256(256);

  // ---- one-time normalization precompute ----
  k_deg_init <<<(N + 255) / 256, b256, 0, stream>>>(dinv, N);
  k_deg_accum<<<(E + 255) / 256, b256, 0, stream>>>(dst, dinv, E);
  k_dinv     <<<(N + 255) / 256, b256, 0, stream>>>(dinv, N);
  k_norm     <<<(E + 255) / 256, b256, 0, stream>>>(src, dst, dinv, normw, E);

  const int elem4 = N * (HDIM / 4);
  const float* Hin = x;
  for (int l = 0; l < L; ++l) {
    const float* Wl = Ws + (size_t)l * HDIM * HDIM;
    k_gemm_wmma<<<(N + 15) / 16, b256, 0, stream>>>(Hin, Wl, Ybuf, N);
    k_init_self<<<(elem4 + 255) / 256, b256, 0, stream>>>(Ybuf, dinv, bs + l * HDIM, Hagg, N);
    k_agg      <<<(E + 7) / 8, b256, 0, stream>>>(Ybuf, src, dst, normw, Hagg, E);
    k_zero_stats<<<1, b256, 0, stream>>>(stats);
    k_stats    <<<(N + 255) / 256, b256, 0, stream>>>(Hagg, stats, N);
    k_finalize <<<1, 128, 0, stream>>>(stats, gammas + l * HDIM, betas + l * HDIM, N);
    const bool last = (l == L - 1);
    float* dest = last ? (float*)d_out : bufA;
    k_apply    <<<(elem4 + 255) / 256, b256, 0, stream>>>(
        Hagg, stats, last ? attn : nullptr, dest, N);
    Hin = bufA;
  }
}